// ScatterRouter_34359739077
// MI455X (gfx1250) — compile-verified
//
#include <hip/hip_runtime.h>
#include <hip/hip_bf16.h>
#include <stdint.h>

// Problem constants (must match reference)
#define T_TOK 131072
#define D_DIM 256
#define P_NUM 8
#define K_TOP 2
#define CAP 49152
#define TOK_PER_BLOCK 256
#define NBLK (T_TOK / TOK_PER_BLOCK)   // 512

// ---- wave32 helpers (CDNA5 is wave32-only) ----
__device__ __forceinline__ uint32_t ballot32(bool b) {
#if __has_builtin(__builtin_amdgcn_ballot_w32)
    return __builtin_amdgcn_ballot_w32(b);
#else
    return (uint32_t)__ballot(b);
#endif
}

__device__ __forceinline__ uint32_t bcast_lane(uint32_t v, int lane /*uniform*/) {
#if __has_builtin(__builtin_amdgcn_readlane)
    return (uint32_t)__builtin_amdgcn_readlane((int)v, lane);
#else
    return (uint32_t)__shfl((int)v, lane, 32);
#endif
}

// ---- gfx1250 async global<->LDS path (ASYNCcnt-tracked) ----
#if __has_builtin(__builtin_amdgcn_global_load_async_to_lds_b128) && \
    __has_builtin(__builtin_amdgcn_global_store_async_from_lds_b128)
#define HAVE_ASYNC_LDS 1
#else
#define HAVE_ASYNC_LDS 0
#endif

#if HAVE_ASYNC_LDS
// Builtin signature (from clang diagnostics): param0 = v4i32 in AS1 (__device__),
// param1 = v4i32 in AS3 (__shared__), then imm offset (ICE), imm cpol.
typedef int v4i __attribute__((vector_size(16)));
typedef __attribute__((address_space(1))) v4i g_v4i;
typedef __attribute__((address_space(3))) v4i l_v4i;
__device__ __forceinline__ g_v4i* asg(uintptr_t a) { return (g_v4i*)a; }
__device__ __forceinline__ l_v4i* asl(uint32_t a)  { return (l_v4i*)(uintptr_t)a; }

__device__ __forceinline__ void wait_async0() {
#if __has_builtin(__builtin_amdgcn_s_wait_asynccnt)
    __builtin_amdgcn_s_wait_asynccnt(0);
#else
    asm volatile("s_wait_asynccnt 0x0" ::: "memory");
#endif
}
#endif

// ---------------------------------------------------------------------------
// Kernel 1: top-2 gating, hot_mask output, packed dests, per-block path counts
// ---------------------------------------------------------------------------
__global__ __launch_bounds__(TOK_PER_BLOCK)
void k_topk(const float* __restrict__ score,
            uint8_t* __restrict__ destpack,
            uint32_t* __restrict__ blockCounts,
            int* __restrict__ hot_mask) {
    __shared__ uint32_t cnt[P_NUM];
    const int tid  = threadIdx.x;
    const int lane = tid & 31;
    const int t    = blockIdx.x * TOK_PER_BLOCK + tid;

    if (tid < P_NUM) cnt[tid] = 0;
    __syncthreads();

    // 8 scores per token, 32B-aligned -> two b128 loads
    const float4* s4 = reinterpret_cast<const float4*>(score + (size_t)t * P_NUM);
    float4 a = s4[0], b = s4[1];
    float v[8] = {a.x, a.y, a.z, a.w, b.x, b.y, b.z, b.w};

    // top_k semantics: ties broken toward lower index -> strict '>' keeps first max
    int d0 = 0; float m0 = v[0];
#pragma unroll
    for (int i = 1; i < 8; ++i) { if (v[i] > m0) { m0 = v[i]; d0 = i; } }
    int d1 = (d0 == 0) ? 1 : 0; float m1 = v[d1];
#pragma unroll
    for (int i = 0; i < 8; ++i) { if (i != d0 && v[i] > m1) { m1 = v[i]; d1 = i; } }

    destpack[t] = (uint8_t)(d0 | (d1 << 4));

    // hot_mask row: int32 0/1, written as two b128 stores
    int hm[8];
#pragma unroll
    for (int p = 0; p < 8; ++p) hm[p] = (p == d0 || p == d1) ? 1 : 0;
    int4* hrow = reinterpret_cast<int4*>(hot_mask + (size_t)t * P_NUM);
    hrow[0] = make_int4(hm[0], hm[1], hm[2], hm[3]);
    hrow[1] = make_int4(hm[4], hm[5], hm[6], hm[7]);

    // per-block per-path counts via wave32 ballots
#pragma unroll
    for (int p = 0; p < 8; ++p) {
        uint32_t m = ballot32(d0 == p || d1 == p);
        if (lane == 0) atomicAdd(&cnt[p], (uint32_t)__popc(m));
    }
    __syncthreads();
    if (tid < P_NUM) blockCounts[blockIdx.x * P_NUM + tid] = cnt[tid];
}

// ---------------------------------------------------------------------------
// Kernel 2: per-path exclusive scan of block counts (one block, 8 waves)
// wave p handles path p; each lane owns 16 consecutive blocks.
// ---------------------------------------------------------------------------
__global__ __launch_bounds__(256)
void k_scan(const uint32_t* __restrict__ blockCounts,
            uint32_t* __restrict__ blockOffs,
            uint32_t* __restrict__ totals) {
    __shared__ uint32_t part[P_NUM][33];
    const int p    = threadIdx.x >> 5;
    const int lane = threadIdx.x & 31;
    const int base = lane * 16;                 // 32 lanes * 16 = 512 = NBLK

    uint32_t vals[16];
    uint32_t sum = 0;
#pragma unroll
    for (int i = 0; i < 16; ++i) {
        vals[i] = blockCounts[(size_t)(base + i) * P_NUM + p];
        sum += vals[i];
    }
    part[p][lane] = sum;
    __syncthreads();

    if (threadIdx.x < P_NUM) {                  // tiny serial scan of 32 lane-partials
        uint32_t run = 0;
#pragma unroll
        for (int l = 0; l < 32; ++l) {
            uint32_t c = part[threadIdx.x][l];
            part[threadIdx.x][l] = run;
            run += c;
        }
        totals[threadIdx.x] = run;
    }
    __syncthreads();

    uint32_t run = part[p][lane];
#pragma unroll
    for (int i = 0; i < 16; ++i) {
        blockOffs[(size_t)(base + i) * P_NUM + p] = run;
        run += vals[i];
    }
}

// ---------------------------------------------------------------------------
// Kernel 3: zero only the unwritten tail rows [total_p, CAP) of each path.
// One wave per output row; 8 rows per block.
// ---------------------------------------------------------------------------
__global__ __launch_bounds__(256)
void k_zero(const uint32_t* __restrict__ totals, float* __restrict__ out) {
    const int w    = threadIdx.x >> 5;
    const int lane = threadIdx.x & 31;
    const int row  = blockIdx.x * 8 + w;        // 0 .. P*CAP-1
    const int p    = row / CAP;
    const int s    = row - p * CAP;
    const uint32_t tot = totals[p];
    if ((uint32_t)s >= tot) {
        float4 z = make_float4(0.f, 0.f, 0.f, 0.f);
        float4* dst = reinterpret_cast<float4*>(out + (size_t)row * D_DIM);
        dst[lane]      = z;
        dst[lane + 32] = z;
    }
}

// ---------------------------------------------------------------------------
// Kernel 4: ordered scatter. Intra-wave prefix by ballot, cross-wave by LDS,
// grid-level by blockOffs. Row copy via gfx1250 async global<->LDS engine
// (ASYNCcnt), falling back to coalesced VGPR b128 copies if unavailable.
// ---------------------------------------------------------------------------
__global__ __launch_bounds__(TOK_PER_BLOCK)
void k_scatter(const float* __restrict__ in_flow,
               const uint8_t* __restrict__ destpack,
               const uint32_t* __restrict__ blockOffs,
               float* __restrict__ out) {
    __shared__ uint32_t sWaveCnt[8 * P_NUM];
    __shared__ uint32_t sBase[8 * P_NUM];
#if HAVE_ASYNC_LDS
    __shared__ __align__(16) float stage[8][4 * D_DIM];   // 4 rows per wave = 32 KB
#endif

    const int tid  = threadIdx.x;
    const int w    = tid >> 5;
    const int lane = tid & 31;
    const int t    = blockIdx.x * TOK_PER_BLOCK + tid;

    const uint32_t pk = destpack[t];
    const int d0 = pk & 7;
    const int d1 = (pk >> 4) & 7;

    const uint32_t below = (1u << lane) - 1u;
    uint32_t pre0 = 0, pre1 = 0;
#pragma unroll
    for (int p = 0; p < 8; ++p) {
        uint32_t m  = ballot32(d0 == p || d1 == p);
        uint32_t pc = __popc(m & below);
        if (d0 == p) pre0 = pc;
        if (d1 == p) pre1 = pc;
        if (lane == 0) sWaveCnt[w * 8 + p] = __popc(m);
    }
    __syncthreads();

    if (tid < P_NUM) {                           // per-path scan across the 8 waves
        uint32_t run = blockOffs[blockIdx.x * P_NUM + tid];
#pragma unroll
        for (int ww = 0; ww < 8; ++ww) {
            sBase[ww * 8 + tid] = run;
            run += sWaveCnt[ww * 8 + tid];
        }
    }
    __syncthreads();

    const uint32_t s0 = sBase[w * 8 + d0] + pre0;   // token-ordered slot in path d0
    const uint32_t s1 = sBase[w * 8 + d1] + pre1;
    const uint32_t flat0 = (s0 < CAP) ? (uint32_t)d0 * CAP + s0 : 0xFFFFFFFFu;
    const uint32_t flat1 = (s1 < CAP) ? (uint32_t)d1 * CAP + s1 : 0xFFFFFFFFu;

    const int tokBase = blockIdx.x * TOK_PER_BLOCK + w * 32;
    const float* srcBase = in_flow + (size_t)tokBase * D_DIM;

    // gfx1250 prefetch of this wave's upcoming rows (global_prefetch_b8)
    __builtin_prefetch(srcBase + (size_t)lane * D_DIM, 0, 3);

#if HAVE_ASYNC_LDS
    // -------- async global->LDS->global row pipeline (no VGPR data path) ----
    const uintptr_t srcLane = (uintptr_t)srcBase + (uintptr_t)lane * 16u;  // 512B rake
    const uintptr_t outAddr = (uintptr_t)out;
    // LDS byte offset of this wave's stage (low 32 bits of the shared address)
    const uint32_t ldsWave = (uint32_t)(uintptr_t)&stage[w][0];
    const uint32_t ldsLane = ldsWave + (uint32_t)lane * 16u;

    for (int g = 0; g < 8; ++g) {                 // 8 groups x 4 rows
        wait_async0();                            // stage free (prev stores drained)
        const uintptr_t gsrc = srcLane + (uintptr_t)g * (4u * 1024u);
        // 4KB in = 8 x 512B async loads; imm offset applies to BOTH global+LDS
#define ASYNC_LD(IMM) \
        __builtin_amdgcn_global_load_async_to_lds_b128(asg(gsrc), asl(ldsLane), (IMM), 0)
        ASYNC_LD(0);    ASYNC_LD(512);  ASYNC_LD(1024); ASYNC_LD(1536);
        ASYNC_LD(2048); ASYNC_LD(2560); ASYNC_LD(3072); ASYNC_LD(3584);
#undef ASYNC_LD
        wait_async0();                            // loads complete
#pragma unroll
        for (int rr = 0; rr < 4; ++rr) {
            const int r = g * 4 + rr;
            const uint32_t f0 = bcast_lane(flat0, r);   // wave-uniform dests
            const uint32_t f1 = bcast_lane(flat1, r);
            const uint32_t lsrc = ldsLane + (unsigned)rr * 1024u;
            if (f0 != 0xFFFFFFFFu) {
                const uintptr_t gd = outAddr + (uintptr_t)f0 * 1024u + (uintptr_t)lane * 16u;
                __builtin_amdgcn_global_store_async_from_lds_b128(asg(gd), asl(lsrc), 0,   0);
                __builtin_amdgcn_global_store_async_from_lds_b128(asg(gd), asl(lsrc), 512, 0);
            }
            if (f1 != 0xFFFFFFFFu) {
                const uintptr_t gd = outAddr + (uintptr_t)f1 * 1024u + (uintptr_t)lane * 16u;
                __builtin_amdgcn_global_store_async_from_lds_b128(asg(gd), asl(lsrc), 0,   0);
                __builtin_amdgcn_global_store_async_from_lds_b128(asg(gd), asl(lsrc), 512, 0);
            }
        }
    }
    wait_async0();                                // drain before wave exit
#else
    // -------- fallback: cooperative coalesced VGPR b128 copies --------------
    for (int r = 0; r < 32; ++r) {
        const uint32_t f0 = bcast_lane(flat0, r);
        const uint32_t f1 = bcast_lane(flat1, r);
        const float4* src = reinterpret_cast<const float4*>(srcBase + (size_t)r * D_DIM);
        float4 x0 = src[lane];
        float4 x1 = src[lane + 32];
        if (f0 != 0xFFFFFFFFu) {
            float4* dst = reinterpret_cast<float4*>(out + (size_t)f0 * D_DIM);
            dst[lane]      = x0;
            dst[lane + 32] = x1;
        }
        if (f1 != 0xFFFFFFFFu) {
            float4* dst = reinterpret_cast<float4*>(out + (size_t)f1 * D_DIM);
            dst[lane]      = x0;
            dst[lane + 32] = x1;
        }
    }
#endif
}

// ---------------------------------------------------------------------------
extern "C" void kernel_launch(void* const* d_in, const int* in_sizes, int n_in,
                              void* d_out, int out_size, void* d_ws, size_t ws_size,
                              hipStream_t stream) {
    const float* in_flow = (const float*)d_in[0];   // (T, D) f32
    const float* score   = (const float*)d_in[1];   // (T, P) f32

    float* out = (float*)d_out;                                   // (P, CAP, D) f32
    int*   hot = (int*)(out + (size_t)P_NUM * CAP * D_DIM);       // (T, P) int32

    // workspace layout (~164 KB)
    uint8_t*  destpack    = (uint8_t*)d_ws;                       // T bytes
    uint32_t* blockCounts = (uint32_t*)((char*)d_ws + T_TOK);     // NBLK*P u32
    uint32_t* blockOffs   = blockCounts + (size_t)NBLK * P_NUM;   // NBLK*P u32
    uint32_t* totals      = blockOffs + (size_t)NBLK * P_NUM;     // P u32

    k_topk   <<<NBLK,            TOK_PER_BLOCK, 0, stream>>>(score, destpack, blockCounts, hot);
    k_scan   <<<1,               256,           0, stream>>>(blockCounts, blockOffs, totals);
    k_zero   <<<P_NUM * CAP / 8, 256,           0, stream>>>(totals, out);
    k_scatter<<<NBLK,            TOK_PER_BLOCK, 0, stream>>>(in_flow, destpack, blockOffs, out);
}